// GraphNet_89773406421119
// MI455X (gfx1250) — compile-verified
//
#include <hip/hip_runtime.h>

// Problem constants (match reference)
#define B_EV   512
#define NPE    512
#define NTOT   (B_EV * NPE)
#define KNN    6
#define F_IN   128
#define H      16
#define OUTF   10
#define GRID   16          // COORD_MAX
#define NCELL  (GRID * GRID)

typedef __attribute__((ext_vector_type(2))) float f32x2;
typedef __attribute__((ext_vector_type(8))) float f32x8;

// D = A(16x4 f32) * B(4x16 f32) + C(16x16 f32), wave32.
static __device__ __forceinline__ f32x8 wmma_f32_16x16x4(f32x2 a, f32x2 b, f32x8 c) {
#if __has_builtin(__builtin_amdgcn_wmma_f32_16x16x4_f32)
  return __builtin_amdgcn_wmma_f32_16x16x4_f32(false, a, false, b, (short)0, c, false, false);
#else
  f32x8 d = c;
  asm volatile("v_wmma_f32_16x16x4_f32 %0, %1, %2, %0"
               : "+v"(d) : "v"(a), "v"(b));
  return d;
#endif
}

__launch_bounds__(NPE)
__global__ void fused_gcn_kernel(const int* __restrict__ coo,
                                 const float* __restrict__ x,
                                 const float* __restrict__ W1,
                                 const float* __restrict__ b1,
                                 const float* __restrict__ W2,
                                 const float* __restrict__ b2,
                                 float* __restrict__ out)
{
  // LDS (~96 KB of 320 KB/WGP): whole event stays on the WGP.
  __shared__ __align__(16) float s_Wt[H][F_IN + 4];   // W1^T, stride 132 -> conflict-free B frags
  __shared__ __align__(16) float s_h [NPE][H + 4];    // h0, then h1 (stride 20)
  __shared__ __align__(16) float s_h2[NPE][H + 4];    // layer-2 pre-aggregation (stride 20)
  __shared__ __align__(16) float s_W2t[H][H + 4];     // W2^T zero-padded to 16x16
  __shared__ int   s_cnt[NCELL];                      // cell histogram
  __shared__ int   s_scan[NCELL];                     // inclusive scan buffer
  __shared__ int   s_start[NCELL + 1];                // cell start offsets
  __shared__ int   s_cur[NCELL];                      // scatter cursors
  __shared__ int   s_pts[NPE];                        // node ids sorted by cell
  __shared__ float s_b1[H];
  __shared__ float s_b2[H];

  const int tid  = threadIdx.x;           // 0..511
  const int ev   = blockIdx.x;
  const int n0   = ev * NPE;
  const int lane = tid & 31;
  const int w    = tid >> 5;              // wave 0..15
  const int lm   = lane & 15;             // M (A) / N col (B,C,D)
  const int kd   = (lane >> 4) << 1;      // K sub-offset: 0 or 2
  const int half = (lane >> 4) << 3;      // C/D row half: 0 or 8

  // ---- stage constants; warm caches for the event's x tile (256 KB) ----
  const int xi = coo[(size_t)(n0 + tid) * 3 + 0];
  const int yi = coo[(size_t)(n0 + tid) * 3 + 1];
  {
    const float* xr = x + (size_t)(n0 + tid) * F_IN;   // 512 B / row
    __builtin_prefetch(xr,      0, 3);                 // -> global_prefetch_b8
    __builtin_prefetch(xr + 32, 0, 3);
    __builtin_prefetch(xr + 64, 0, 3);
    __builtin_prefetch(xr + 96, 0, 3);
  }
  for (int idx = tid; idx < F_IN * H; idx += NPE) {    // 4 iters
    int k = idx >> 4, c = idx & 15;
    s_Wt[c][k] = W1[idx];
  }
  if (tid < H * H) {
    int o = tid >> 4, k = tid & 15;
    s_W2t[o][k] = (o < OUTF) ? W2[k * OUTF + o] : 0.0f;
  }
  if (tid < H) {
    s_b1[tid] = b1[tid];
    s_b2[tid] = (tid < OUTF) ? b2[tid] : 0.0f;
  }
  if (tid < NCELL) s_cnt[tid] = 0;
  __syncthreads();

  // ---- KNN build: counting-sort nodes by cell (coords ARE cell centers) ----
  const int mycell = yi * GRID + xi;
  atomicAdd(&s_cnt[mycell], 1);
  __syncthreads();
  if (tid < NCELL) s_scan[tid] = s_cnt[tid];
  __syncthreads();
  for (int off = 1; off < NCELL; off <<= 1) {          // Hillis-Steele, 8 steps
    int v = 0;
    if (tid < NCELL) {
      v = s_scan[tid];
      if (tid >= off) v += s_scan[tid - off];
    }
    __syncthreads();
    if (tid < NCELL) s_scan[tid] = v;
    __syncthreads();
  }
  if (tid < NCELL) s_start[tid + 1] = s_scan[tid];
  if (tid == 0)    s_start[0] = 0;
  __syncthreads();
  if (tid < NCELL) s_cur[tid] = s_start[tid];
  __syncthreads();
  {
    int slot = atomicAdd(&s_cur[mycell], 1);
    s_pts[slot] = tid;                                 // scatter order irrelevant: selection is a set
  }
  __syncthreads();

  // ---- layer 1 GEMM: h0 = x @ W1 via v_wmma_f32_16x16x4_f32 ----
  #pragma unroll
  for (int t = 0; t < 2; ++t) {
    const int r0 = w * 32 + t * 16;
    const float* xp = x + (size_t)(n0 + r0 + lm) * F_IN + kd;
    f32x8 acc = {0.f, 0.f, 0.f, 0.f, 0.f, 0.f, 0.f, 0.f};
    #pragma unroll
    for (int kk = 0; kk < F_IN; kk += 4) {
      f32x2 a  = *(const f32x2*)(xp + kk);
      f32x2 bf = *(const f32x2*)&s_Wt[lm][kk + kd];
      acc = wmma_f32_16x16x4(a, bf, acc);
    }
    #pragma unroll
    for (int v = 0; v < 8; ++v)
      s_h[r0 + half + v][lm] = acc[v];                 // C/D: M = half+v, N = lm
  }

  // ---- KNN query: expanding Chebyshev rings; key = (d^2<<9)|j == top_k order ----
  unsigned best[KNN];
  #pragma unroll
  for (int e = 0; e < KNN; ++e) best[e] = 0xFFFFFFFFu;

  auto insert_key = [&](unsigned key) {
    if (key < best[KNN - 1]) {
      unsigned cur = key;
      #pragma unroll
      for (int e2 = 0; e2 < KNN; ++e2) {
        unsigned mn = best[e2] < cur ? best[e2] : cur;
        cur         = best[e2] < cur ? cur : best[e2];
        best[e2] = mn;
      }
    }
  };
  auto proc_cell = [&](int cx, int cy) {
    int cell = cy * GRID + cx;
    int pb = s_start[cell], pe = s_start[cell + 1];
    int ddx = xi - cx, ddy = yi - cy;
    unsigned bk = (unsigned)(ddx * ddx + ddy * ddy) << 9;
    for (int p = pb; p < pe; ++p)
      insert_key(bk | (unsigned)s_pts[p]);
  };

  {   // r = 0: own cell (d = 0), exclude self
    int pb = s_start[mycell], pe = s_start[mycell + 1];
    for (int p = pb; p < pe; ++p) {
      int jj = s_pts[p];
      if (jj != tid) insert_key((unsigned)jj);
    }
  }
  for (int r = 1; r < GRID; ++r) {
    unsigned d5 = best[KNN - 1] >> 9;                  // huge while unfilled
    if ((unsigned)(r * r) > d5) break;                 // no remaining cell can beat best[5]
    int xlo = xi - r, xhi = xi + r, ylo = yi - r, yhi = yi + r;
    int cx0 = xlo > 0 ? xlo : 0, cx1 = xhi < GRID - 1 ? xhi : GRID - 1;
    for (int cx = cx0; cx <= cx1; ++cx) {
      if (ylo >= 0)   proc_cell(cx, ylo);
      if (yhi < GRID) proc_cell(cx, yhi);
    }
    int cy0 = (ylo + 1) > 0 ? (ylo + 1) : 0, cy1 = (yhi - 1) < GRID - 1 ? (yhi - 1) : GRID - 1;
    for (int cy = cy0; cy <= cy1; ++cy) {
      if (xlo >= 0)   proc_cell(xlo, cy);
      if (xhi < GRID) proc_cell(xhi, cy);
    }
  }
  __syncthreads();   // s_h (h0) complete for whole event

  // ---- aggregation 1: h1 = relu((sum_nbr h0 + self)/7 + b1)  (deg == 7) ----
  float h1r[H];
  #pragma unroll
  for (int c = 0; c < H; c += 2) {
    f32x2 v = *(const f32x2*)&s_h[tid][c];
    h1r[c] = v.x; h1r[c + 1] = v.y;
  }
  #pragma unroll
  for (int e = 0; e < KNN; ++e) {
    int j = (int)(best[e] & 511u);
    #pragma unroll
    for (int c = 0; c < H; c += 2) {
      f32x2 v = *(const f32x2*)&s_h[j][c];
      h1r[c] += v.x; h1r[c + 1] += v.y;
    }
  }
  const float inv7 = 1.0f / 7.0f;
  #pragma unroll
  for (int c = 0; c < H; ++c)
    h1r[c] = fmaxf(h1r[c] * inv7 + s_b1[c], 0.0f);

  __syncthreads();   // everyone done reading h0 before overwrite
  #pragma unroll
  for (int c = 0; c < H; c += 2) {
    f32x2 v; v.x = h1r[c]; v.y = h1r[c + 1];
    *(f32x2*)&s_h[tid][c] = v;                         // s_h now holds h1
  }
  __syncthreads();

  // ---- layer 2 GEMM: h2 = h1 @ W2 (zero-padded 16x16) via WMMA ----
  #pragma unroll
  for (int t = 0; t < 2; ++t) {
    const int r0 = w * 32 + t * 16;
    f32x8 acc = {0.f, 0.f, 0.f, 0.f, 0.f, 0.f, 0.f, 0.f};
    #pragma unroll
    for (int kk = 0; kk < H; kk += 4) {
      f32x2 a  = *(const f32x2*)&s_h[r0 + lm][kk + kd];
      f32x2 bf = *(const f32x2*)&s_W2t[lm][kk + kd];
      acc = wmma_f32_16x16x4(a, bf, acc);
    }
    #pragma unroll
    for (int v = 0; v < 8; ++v)
      s_h2[r0 + half + v][lm] = acc[v];
  }
  __syncthreads();

  // ---- aggregation 2 + bias -> global out [N, 10] ----
  float o2[OUTF];
  #pragma unroll
  for (int c = 0; c < OUTF; c += 2) {
    f32x2 v = *(const f32x2*)&s_h2[tid][c];
    o2[c] = v.x; o2[c + 1] = v.y;
  }
  #pragma unroll
  for (int e = 0; e < KNN; ++e) {
    int j = (int)(best[e] & 511u);
    #pragma unroll
    for (int c = 0; c < OUTF; c += 2) {
      f32x2 v = *(const f32x2*)&s_h2[j][c];
      o2[c] += v.x; o2[c + 1] += v.y;
    }
  }
  float* op = out + (size_t)(n0 + tid) * OUTF;         // 40-byte rows, 8B aligned
  #pragma unroll
  for (int c = 0; c < OUTF; c += 2) {
    f32x2 v;
    v.x = o2[c]     * inv7 + s_b2[c];
    v.y = o2[c + 1] * inv7 + s_b2[c + 1];
    *(f32x2*)(op + c) = v;
  }
}

extern "C" void kernel_launch(void* const* d_in, const int* in_sizes, int n_in,
                              void* d_out, int out_size, void* d_ws, size_t ws_size,
                              hipStream_t stream) {
  (void)in_sizes; (void)n_in; (void)out_size; (void)d_ws; (void)ws_size;
  const int*   coo = (const int*)  d_in[0];
  const float* x   = (const float*)d_in[1];
  const float* W1  = (const float*)d_in[2];
  const float* b1  = (const float*)d_in[3];
  const float* W2  = (const float*)d_in[4];
  const float* b2  = (const float*)d_in[5];
  float* out = (float*)d_out;
  fused_gcn_kernel<<<B_EV, NPE, 0, stream>>>(coo, x, W1, b1, W2, b2, out);
}